// SparseExpertV2_63136019251705
// MI455X (gfx1250) — compile-verified
//
#include <hip/hip_runtime.h>
#include <hip/hip_bf16.h>
#include <math.h>

// Problem constants (from reference): D=1024, M=4096, B=8, K=8, K_DRAFT=16
#define DD   1024
#define MM   4096
#define BB   8
#define MB   32768      // M*B
#define NN   2048       // 2*1024 tokens
#define KDR  16
#define KK   8
#define OUT_ELEMS 2097152  // 2*1024*1024

typedef __attribute__((ext_vector_type(2))) float v2f;
typedef __attribute__((ext_vector_type(8))) float v8f;
typedef unsigned int __attribute__((ext_vector_type(4))) u32x4;
typedef int          __attribute__((ext_vector_type(4))) i32x4;
typedef int          __attribute__((ext_vector_type(8))) i32x8;

__device__ __forceinline__ v8f wmma4(v2f a, v2f b, v8f c) {
  // V_WMMA_F32_16X16X4_F32 : D = A(16x4) * B(4x16) + C(16x16), all fp32
  return __builtin_amdgcn_wmma_f32_16x16x4_f32(
      /*neg_a=*/false, a, /*neg_b=*/false, b,
      /*c_mod=*/(short)0, c, /*reuse_a=*/false, /*reuse_b=*/false);
}

// ---------------------------------------------------------------------------
// TDM helper: issue a 2-D tile DMA (global -> LDS) via the Tensor Data Mover.
// D# built per CDNA5 ISA §8.3/8.4:
//   group0: [1:0]count=1, [63:32]lds_addr, [120:64]global_addr, [127:126]type=2
//   group1: w0 = data_size(2=4B)<<16 | pad_enable<<20 | pad_interval<<22
//                | pad_amount<<25   (workgroup_mask[15:0]=0: not in cluster)
//           w1 = tensor_dim0[15:0]<<16
//           w2 = tensor_dim0[31:16] | tensor_dim1[15:0]<<16
//           w3 = tensor_dim1[31:16] | tile_dim0<<16
//           w4 = tile_dim1 (tile_dim2=0)
//           w5 = tensor_dim0_stride[31:0]  (w6=w7=0)
//   groups 2/3 (and trailing group): zero (2-D tile, tile_dim3/4 unused)
// This toolchain exposes the 6-arg builtin:
//   (uint32x4 g0, int32x8 g1, int32x4, int32x4, int32x8, i32 cpol)
// ---------------------------------------------------------------------------
__device__ __forceinline__ void tdm_load_2d(const void* gaddr, unsigned lds_off,
                                            unsigned w0, unsigned w1, unsigned w2,
                                            unsigned w3, unsigned w4, unsigned w5) {
  const unsigned long long ga = (unsigned long long)gaddr;
  u32x4 g0;
  g0[0] = 1u;                                             // count=1, user D#
  g0[1] = lds_off;                                        // LDS byte address
  g0[2] = (unsigned)(ga & 0xffffffffull);                 // global_addr[31:0]
  g0[3] = (unsigned)((ga >> 32) & 0x1ffffffull) | (2u << 30); // addr[56:32]|type=2
  i32x8 g1;
  g1[0] = (int)w0; g1[1] = (int)w1; g1[2] = (int)w2; g1[3] = (int)w3;
  g1[4] = (int)w4; g1[5] = (int)w5; g1[6] = 0; g1[7] = 0;
  i32x4 z4; z4[0] = 0; z4[1] = 0; z4[2] = 0; z4[3] = 0;
  i32x8 z8; z8[0] = 0; z8[1] = 0; z8[2] = 0; z8[3] = 0;
  z8[4] = 0; z8[5] = 0; z8[6] = 0; z8[7] = 0;
  __builtin_amdgcn_tensor_load_to_lds(g0, g1, z4, z4, z8, 0);
}

__device__ __forceinline__ unsigned lds_offset(const void* p) {
  // generic LDS pointer: aperture in high bits, byte offset in addr[31:0]
  return (unsigned)(unsigned long long)p;
}

// ---------------------------------------------------------------------------
// Kernel 1: vinv[m*8+b] = 1 / ||V[:,m,b]||   (norm over D, coalesced over mb)
// ---------------------------------------------------------------------------
__global__ __launch_bounds__(256) void vnorm_kernel(const float* __restrict__ V,
                                                    float* __restrict__ vinv) {
  const int mb = blockIdx.x * 256 + threadIdx.x;   // 0..32767
  float a = 0.f;
  for (int d = 0; d < DD; ++d) {
    float v = V[d * MB + mb];
    a += v * v;
  }
  vinv[mb] = rsqrtf(a);
}

// ---------------------------------------------------------------------------
// Kernel 2: uscale[m*8+b] = tanh(u_scales[m,b]) / ||U[m,:,b]||
// ---------------------------------------------------------------------------
__global__ __launch_bounds__(256) void uscale_kernel(const float* __restrict__ U,
                                                     const float* __restrict__ us,
                                                     float* __restrict__ uscale) {
  __shared__ float sred[256];
  const int m = blockIdx.x, tid = threadIdx.x;
  float a = 0.f;
  for (int e = tid; e < DD * BB; e += 256) {   // e = d*8+b ; (e&7)==(tid&7)
    float v = U[m * DD * BB + e];
    a += v * v;
  }
  sred[tid] = a;
  __syncthreads();
  for (int off = 128; off >= 8; off >>= 1) {
    if (tid < off) sred[tid] += sred[tid + off];
    __syncthreads();
  }
  if (tid < 8)
    uscale[m * 8 + tid] = tanhf(us[m * 8 + tid]) * rsqrtf(sred[tid]);
}

// ---------------------------------------------------------------------------
// Kernel 3: energy[n,m] = sum_b ( (x[n,:] . V[:,m,b]) * vinv[m,b] )^2
// WMMA f32 GEMM, TDM-staged + double-buffered LDS.
//   Block tile: 32 rows(n) x 128 cols(mb); 8 waves as 2x4, each wave 16x32
//   (two accumulators sharing one A fragment). K chunked by 32.
//   Wave0 DMAs the V tile (rows padded to 136 floats by the TDM pad feature),
//   wave1 DMAs the x tile (rows padded to 33 floats). s_wait_tensorcnt <=1
//   after issuing chunk c+1 retires chunk c (TENSORcnt is per-wave, in-order).
// ---------------------------------------------------------------------------
#define VS_STRIDE 136   // 128 data + 8 pad DWORDs  (pad_interval=6, pad_amount=7)
#define XS_STRIDE 33    // 32 data + 1 pad DWORD    (pad_interval=4, pad_amount=0)

__global__ __launch_bounds__(256) void energy_kernel(const float* __restrict__ x,
                                                     const float* __restrict__ V,
                                                     const float* __restrict__ vinv,
                                                     float* __restrict__ energy) {
  __shared__ float xs[2][32 * XS_STRIDE];
  __shared__ float vs[2][32 * VS_STRIDE];
  const int tid  = threadIdx.x;
  const int lane = tid & 31;
  const int w    = tid >> 5;
  const int wn   = w >> 2, wc = w & 3;
  const int lm   = lane & 15, half = lane >> 4;
  const int rb   = blockIdx.y * 32;    // token-row base
  const int cb   = blockIdx.x * 128;   // mb-col base

  // group1 words (compile-time constants)
  //   V tile: 32(k) x 128(mb), tensor 1024 x 32768, stride 32768, pad 128+8
  const unsigned vw0 = (2u << 16) | (1u << 20) | (6u << 22) | (7u << 25);
  const unsigned vw1 = (MB & 0xffffu) << 16;
  const unsigned vw2 = (MB >> 16) | ((DD & 0xffffu) << 16);
  const unsigned vw3 = (DD >> 16) | (128u << 16);
  //   x tile: 32(n) x 32(k), tensor 2048 x 1024, stride 1024, pad 32+1
  const unsigned xw0 = (2u << 16) | (1u << 20) | (4u << 22);
  const unsigned xw1 = (DD & 0xffffu) << 16;
  const unsigned xw2 = (DD >> 16) | ((NN & 0xffffu) << 16);
  const unsigned xw3 = (NN >> 16) | (32u << 16);

  // prologue: DMA chunk 0
  if (w == 0) {
    tdm_load_2d(V + (size_t)0 * MB + cb, lds_offset(&vs[0][0]),
                vw0, vw1, vw2, vw3, 32u, (unsigned)MB);
  } else if (w == 1) {
    tdm_load_2d(x + (size_t)rb * DD + 0, lds_offset(&xs[0][0]),
                xw0, xw1, xw2, xw3, 32u, (unsigned)DD);
  }

  v8f acc0 = {}, acc1 = {};
  for (int c = 0; c < 32; ++c) {      // 32 K-chunks of 32
    const int cur = c & 1;
    if (c < 31) {
      const int k1 = (c + 1) * 32;
      if (w == 0) {
        tdm_load_2d(V + (size_t)k1 * MB + cb, lds_offset(&vs[cur ^ 1][0]),
                    vw0, vw1, vw2, vw3, 32u, (unsigned)MB);
      } else if (w == 1) {
        tdm_load_2d(x + (size_t)rb * DD + k1, lds_offset(&xs[cur ^ 1][0]),
                    xw0, xw1, xw2, xw3, 32u, (unsigned)DD);
      }
      if (w < 2) __builtin_amdgcn_s_wait_tensorcnt(1);  // chunk c retired
    } else {
      if (w < 2) __builtin_amdgcn_s_wait_tensorcnt(0);
    }
    __syncthreads();

    const float* __restrict__ xsb = &xs[cur][0];
    const float* __restrict__ vsb = &vs[cur][0];
#pragma unroll
    for (int kk = 0; kk < 32; kk += 4) {
      const int ka = kk + 2 * half;
      v2f a, b0, b1;
      a.x  = xsb[(wn * 16 + lm) * XS_STRIDE + ka];
      a.y  = xsb[(wn * 16 + lm) * XS_STRIDE + ka + 1];
      b0.x = vsb[ka * VS_STRIDE + wc * 32 + lm];
      b0.y = vsb[(ka + 1) * VS_STRIDE + wc * 32 + lm];
      b1.x = vsb[ka * VS_STRIDE + wc * 32 + 16 + lm];
      b1.y = vsb[(ka + 1) * VS_STRIDE + wc * 32 + 16 + lm];
      acc0 = wmma4(a, b0, acc0);
      acc1 = wmma4(a, b1, acc1);
    }
    __syncthreads();
  }

  // scale by vinv, square, reduce over b (8 consecutive cols = 8 lanes)
#pragma unroll
  for (int t = 0; t < 2; ++t) {
    const int col = cb + wc * 32 + t * 16 + lm;
    const float s = vinv[col];
#pragma unroll
    for (int i = 0; i < 8; ++i) {
      float v = (t == 0 ? acc0[i] : acc1[i]) * s;
      float r = v * v;
      r += __shfl_xor(r, 1, 32);
      r += __shfl_xor(r, 2, 32);
      r += __shfl_xor(r, 4, 32);
      if ((lane & 7) == 0) {
        energy[(rb + wn * 16 + i + 8 * half) * MM + (col >> 3)] = r;
      }
    }
  }
}

// ---------------------------------------------------------------------------
// Kernel 4: per-token top-16 (value desc, lower index on ties) over M=4096
// ---------------------------------------------------------------------------
__global__ __launch_bounds__(256) void top16_kernel(const float* __restrict__ energy,
                                                    int* __restrict__ didx,
                                                    float* __restrict__ dval) {
  __shared__ float ev[MM];
  __shared__ float sval[256];
  __shared__ int   sidx[256];
  const int n = blockIdx.x, tid = threadIdx.x;
  for (int e = tid; e < MM; e += 256) ev[e] = energy[n * MM + e];
  __syncthreads();
  for (int r = 0; r < KDR; ++r) {
    float bv = -3.4e38f; int bi = 0x7fffffff;
    for (int e = tid; e < MM; e += 256) {
      float v = ev[e];
      if (v > bv) { bv = v; bi = e; }
    }
    sval[tid] = bv; sidx[tid] = bi;
    __syncthreads();
    for (int off = 128; off > 0; off >>= 1) {
      if (tid < off) {
        float ov = sval[tid + off]; int oi = sidx[tid + off];
        if (ov > sval[tid] || (ov == sval[tid] && oi < sidx[tid])) {
          sval[tid] = ov; sidx[tid] = oi;
        }
      }
      __syncthreads();
    }
    if (tid == 0) {
      didx[n * KDR + r] = sidx[0];
      dval[n * KDR + r] = sval[0];
      ev[sidx[0]] = -3.4e38f;
    }
    __syncthreads();
  }
}

// ---------------------------------------------------------------------------
// Kernel 5: zero the two scalar accumulators
// ---------------------------------------------------------------------------
__global__ void init_accum(float* accum) { accum[0] = 0.f; accum[1] = 0.f; }

// ---------------------------------------------------------------------------
// Kernel 6: per-token finish.
//   G(16x16) = (1/8) * Vsel_n^T Vsel_n via self-WMMA (A frag == B frag),
//   8 waves split K=8192, LDS-reduce partial tiles; inhibition; top-8;
//   recompute reprs_topk; fused U-write + residual; atomic scalar partials.
// ---------------------------------------------------------------------------
__global__ __launch_bounds__(256) void finish_kernel(const float* __restrict__ x,
                                                     const float* __restrict__ V,
                                                     const float* __restrict__ U,
                                                     const float* __restrict__ vinv,
                                                     const float* __restrict__ uscale,
                                                     const int* __restrict__ didx,
                                                     const float* __restrict__ dval,
                                                     float* __restrict__ out,
                                                     float* __restrict__ accum) {
  __shared__ int   md[16];
  __shared__ float ev16[16], E16[16], corr[16];
  __shared__ float vinvs[128];     // [draft_i][b]
  __shared__ float gpart[8 * 256];
  __shared__ float G[256];
  __shared__ int   fil[8];
  __shared__ float rk[64];         // reprs_topk [k][b]
  __shared__ float uw[64];         // uscale*rk  [k][b]
  __shared__ float partial[256];

  const int n = blockIdx.x, tid = threadIdx.x;

  if (tid < 16) {
    md[tid]   = didx[n * KDR + tid];
    float v   = dval[n * KDR + tid];
    ev16[tid] = v;
    E16[tid]  = sqrtf(fmaxf(v, 0.f));
  }
  __syncthreads();
  if (tid < 128) vinvs[tid] = vinv[md[tid >> 3] * 8 + (tid & 7)];
  __syncthreads();

  // --- G via self-WMMA: lane l holds Vn[k, m_{l%16}] -> valid A AND B frag ---
  const int w = tid >> 5, lane = tid & 31, lm = lane & 15, half = lane >> 4;
  const int mcol = md[lm];
  v8f cacc = {};
  const int kb = w * 1024 + 2 * half;   // flat k = d*8+b
  for (int k0 = 0; k0 < 1024; k0 += 4) {
    const int k  = kb + k0;
    const int d0 = k >> 3, b0 = k & 7;
    const int d1 = (k + 1) >> 3, b1 = (k + 1) & 7;
    v2f f;
    f.x = V[d0 * MB + mcol * 8 + b0] * vinvs[lm * 8 + b0];
    f.y = V[d1 * MB + mcol * 8 + b1] * vinvs[lm * 8 + b1];
    cacc = wmma4(f, f, cacc);
  }
#pragma unroll
  for (int i = 0; i < 8; ++i)
    gpart[w * 256 + (i + 8 * half) * 16 + lm] = cacc[i];
  __syncthreads();
  {
    float g = 0.f;
#pragma unroll
    for (int w2 = 0; w2 < 8; ++w2) g += gpart[w2 * 256 + tid];
    G[tid] = g * 0.125f;   // / (||Vflat_i|| ||Vflat_j||) = 1/8
  }
  __syncthreads();

  // --- inhibition & corrected ---
  if (tid < 16) {
    float inh = 0.f;
    for (int i = 0; i < 16; ++i)
      if (i != tid) inh += E16[i] * G[i * 16 + tid];
    corr[tid] = ev16[tid] - inh * E16[tid];
  }
  __syncthreads();

  // --- top-8 of corrected (serial, 16 elems) ---
  if (tid == 0) {
    float tmp[16];
    for (int i = 0; i < 16; ++i) tmp[i] = corr[i];
    for (int r = 0; r < KK; ++r) {
      float bv = tmp[0]; int bi = 0;
      for (int i = 1; i < 16; ++i)
        if (tmp[i] > bv) { bv = tmp[i]; bi = i; }
      fil[r] = bi;
      tmp[bi] = -3.4e38f;
    }
  }
  __syncthreads();

  // --- recompute reprs_topk[k][b] = vinv * (x[n,:] . V[:,m_k,b]) ---
  {
    const int p = tid >> 2, sub = tid & 3;   // p in 0..63 -> (k,b)
    const int kk2 = p >> 3, b = p & 7;
    const int mm = md[fil[kk2]];
    const float* xr = x + n * DD;
    float a = 0.f;
    const int d0 = sub * 256;
    for (int d = d0; d < d0 + 256; ++d)
      a += xr[d] * V[d * MB + mm * 8 + b];
    partial[tid] = a;
  }
  __syncthreads();
  if (tid < 64) {
    float s = partial[tid * 4] + partial[tid * 4 + 1] +
              partial[tid * 4 + 2] + partial[tid * 4 + 3];
    rk[tid] = s * vinvs[fil[tid >> 3] * 8 + (tid & 7)];
  }
  __syncthreads();
  if (tid < 64)
    uw[tid] = uscale[md[fil[tid >> 3]] * 8 + (tid & 7)] * rk[tid];
  __syncthreads();

  // --- out[n,d] = x[n,d] + sum_k sum_b U[m_k,d,b]*uw[k][b] ---
  for (int d = tid; d < DD; d += 256) {
    float a = x[n * DD + d];
#pragma unroll
    for (int kk2 = 0; kk2 < KK; ++kk2) {
      const float* up = U + md[fil[kk2]] * (DD * BB) + d * 8;
#pragma unroll
      for (int b = 0; b < 8; ++b) a += up[b] * uw[kk2 * 8 + b];
    }
    out[n * DD + d] = a;
  }

  // --- scalar partials ---
  if (tid == 0) {
    float raw = 0.f, ov = 0.f;
    for (int i = 0; i < KK; ++i) {
      float Ei = E16[fil[i]];
      raw += Ei * Ei;
      for (int j = 0; j < KK; ++j)
        if (j != i) ov += Ei * G[fil[i] * 16 + fil[j]] * E16[fil[j]];
    }
    atomicAdd(&accum[0], raw);
    atomicAdd(&accum[1], ov);
  }
}

// ---------------------------------------------------------------------------
// Kernel 7: scalar epilogue
// ---------------------------------------------------------------------------
__global__ void scalars_kernel(const float* __restrict__ accum,
                               float* __restrict__ out) {
  const float raw = accum[0], ov = accum[1];
  const float total = (raw - ov) / (float)NN;
  out[OUT_ELEMS + 0] = total;
  out[OUT_ELEMS + 1] = raw / (float)NN;
  out[OUT_ELEMS + 2] = ov / (float)NN;
  out[OUT_ELEMS + 3] = -total;
}

// ---------------------------------------------------------------------------
extern "C" void kernel_launch(void* const* d_in, const int* in_sizes, int n_in,
                              void* d_out, int out_size, void* d_ws, size_t ws_size,
                              hipStream_t stream) {
  const float* x   = (const float*)d_in[0];   // (2,1024,1024)
  const float* V   = (const float*)d_in[1];   // (1024,4096,8)
  const float* U   = (const float*)d_in[2];   // (4096,1024,8)
  const float* us  = (const float*)d_in[3];   // (4096,8)
  float* out = (float*)d_out;

  float* ws = (float*)d_ws;
  float* vinv     = ws;                         // 32768
  float* uscale   = ws + 32768;                 // 32768
  float* accum    = ws + 65536;                 // 2 (+pad)
  float* dval     = ws + 65552;                 // 2048*16
  int*   didx     = (int*)(ws + 98320);         // 2048*16
  float* energy   = ws + 131088;                // 2048*4096 (32 MB)

  vnorm_kernel<<<MB / 256, 256, 0, stream>>>(V, vinv);
  uscale_kernel<<<MM, 256, 0, stream>>>(U, us, uscale);
  energy_kernel<<<dim3(MB / 128, NN / 32), 256, 0, stream>>>(x, V, vinv, energy);
  top16_kernel<<<NN, 256, 0, stream>>>(energy, didx, dval);
  init_accum<<<1, 1, 0, stream>>>(accum);
  finish_kernel<<<NN, 256, 0, stream>>>(x, V, U, vinv, uscale, didx, dval, out, accum);
  scalars_kernel<<<1, 1, 0, stream>>>(accum, out);
}